// WaveletMixing_67568425500833
// MI455X (gfx1250) — compile-verified
//
#include <hip/hip_runtime.h>
#include <stdint.h>

// ---------------- DB4 filter constants ----------------
#define DL0 (-0.010597401784997278f)
#define DL1 ( 0.032883011666982945f)
#define DL2 ( 0.030841381835986965f)
#define DL3 (-0.18703481171888114f)
#define DL4 (-0.02798376941698385f)
#define DL5 ( 0.6308807679295904f)
#define DL6 ( 0.7148465705525415f)
#define DL7 ( 0.23037781330885523f)

// Analysis correlation filters (XLA conv = correlation; reference pre-reverses):
//   a[i] = sum_k GLO[k] * x[2i-6+k],  d[i] = sum_k GHI[k] * x[2i-6+k]
constexpr float F_GLO[8] = { DL7,  DL6,  DL5,  DL4,  DL3,  DL2,  DL1,  DL0};
constexpr float F_GHI[8] = { DL0, -DL1,  DL2, -DL3,  DL4, -DL5,  DL6, -DL7};
// Synthesis (transposed conv, lhs_dilation=2, pad 1):
//   r[t] = sum_{k == (1-t) mod 2} SLO[k]*a[(t+k-1)/2] + SHI[k]*d[(t+k-1)/2]
constexpr float F_SLO[8] = { DL0,  DL1,  DL2,  DL3,  DL4,  DL5,  DL6,  DL7};
constexpr float F_SHI[8] = {-DL7,  DL6, -DL5,  DL4, -DL3,  DL2, -DL1,  DL0};

// ---------------- problem / tiling constants ----------------
constexpr int NB    = 8;      // batch
constexpr int S     = 4096;   // sequence
constexpr int H     = 768;    // heads
constexpr int COLS  = 16;     // h-columns per block (= one WMMA N tile)
constexpr int CHUNK = 128;    // kept samples per chunk
constexpr int HALO  = 64;     // halo each side (>= 3-level support ~56)
constexpr int EXT   = CHUNK + 2*HALO;            // 256
constexpr int GUARD = 8;      // guard rows below input tile (lowest access s=-6)
constexpr int M1    = (EXT + 7) / 2;             // 131
constexpr int M2    = (M1  + 7) / 2;             // 69
constexpr int M3    = (M2  + 7) / 2;             // 38
constexpr int R3L   = 2*M3 - 6;                  // 70
constexpr int R2L   = 2*M2 - 6;                  // 132
constexpr int R1L   = 2*M1 - 6;                  // 256 == EXT
constexpr int NT    = 256;                       // threads per block

// LDS layout (units: floats-per-column; buffer reuse: A1 hosts r2, A2 hosts r3)
constexpr int OFF_IN = 0;                        // GUARD + 256
constexpr int OFF_A1 = OFF_IN + GUARD + EXT;     // 132 (a1 then r2)
constexpr int OFF_D1 = OFF_A1 + R2L;             // 131
constexpr int OFF_A2 = OFF_D1 + M1;              // 70  (a2 then r3)
constexpr int OFF_D2 = OFF_A2 + R3L;             // 69
constexpr int OFF_A3 = OFF_D2 + M2;              // 38
constexpr int OFF_D3 = OFF_A3 + M3;              // 38
constexpr int OFF_R1 = OFF_D3 + M3;              // 256
constexpr int LDS_F  = OFF_R1 + R1L;             // 998 floats/col -> 63872 B

typedef float v2f __attribute__((ext_vector_type(2)));
typedef float v8f __attribute__((ext_vector_type(8)));

// ---- compile-time A-fragment tables for the banded-GEMM level-1 -----------
// Per ISA A-layout (16x4 f32): lane L holds A[M=L%16, K=kb+{0,1}], kb=(L>=16)?2:0.
// A[mi, j] = g[j - 2*mi] (8-tap band), j = kt*4 + k_local, kt = 0..9 (K=40).
struct alignas(8) f2 { float x, y; };
struct ATab { f2 v[10][32]; };

constexpr ATab make_atab(const float (&g)[8]) {
  ATab t{};
  for (int kt = 0; kt < 10; ++kt)
    for (int lane = 0; lane < 32; ++lane) {
      const int mi = lane & 15;
      const int kb = (lane >> 4) * 2;
      const int t0 = kt * 4 + kb - 2 * mi;
      const int t1 = t0 + 1;
      t.v[kt][lane].x = (t0 >= 0 && t0 < 8) ? g[t0] : 0.0f;
      t.v[kt][lane].y = (t1 >= 0 && t1 < 8) ? g[t1] : 0.0f;
    }
  return t;
}
constexpr ATab TAB_LO = make_atab(F_GLO);
constexpr ATab TAB_HI = make_atab(F_GHI);

__device__ __forceinline__ int reflect(int i, int n) {
  i = (i < 0) ? (-i - 1) : i;
  i = (i >= n) ? (2*n - 1 - i) : i;
  return i;
}

// CDNA5 async global->LDS DMA (ASYNCcnt path), 16B per lane
__device__ __forceinline__ void async_ld_b128(uint32_t ldsAddr, const float* g) {
  asm volatile("global_load_async_to_lds_b128 %0, %1, off"
               :: "v"(ldsAddr), "v"((uint64_t)(uintptr_t)g)
               : "memory");
}

__device__ __forceinline__ void wait_async0() {
#if __has_builtin(__builtin_amdgcn_s_wait_asynccnt)
  __builtin_amdgcn_s_wait_asynccnt(0);
#else
  asm volatile("s_wait_asynccnt 0" ::: "memory");
#endif
}

__global__ __launch_bounds__(NT)
void wavelet_mix_kernel(const float* __restrict__ x,
                        const float* __restrict__ wA,
                        const float* __restrict__ wD,
                        float* __restrict__ out) {
  __shared__ __align__(16) float sm[LDS_F * COLS];

  const int tid   = threadIdx.x;
  const int chunk = blockIdx.x;            // 0..31
  const int h0    = blockIdx.y * COLS;     // 0..752
  const int b     = blockIdx.z;            // 0..7
  const int sbase = chunk * CHUNK - HALO;  // may be negative

  // ---- Stage 0: async DMA global -> LDS (EXT x COLS), symmetric reflect at row edges
#pragma unroll
  for (int t = 0; t < (EXT * COLS) / 4 / NT; ++t) {   // 4 iters
    int idx4 = tid + t * NT;
    int e    = idx4 >> 2;
    int c4   = (idx4 & 3) * 4;
    int sg   = reflect(sbase + e, S);
    const float* gp = x + ((size_t)b * S + sg) * H + h0 + c4;
    uint32_t loff = (uint32_t)(uintptr_t)&sm[(OFF_IN + GUARD + e) * COLS + c4];
    async_ld_b128(loff, gp);
  }
  wait_async0();
  __syncthreads();

  // ---- Stage 1: level-1 analysis as banded GEMM on v_wmma_f32_16x16x4_f32
  // D[mi,n] = sum_j W[mi,j] * in[2*m0 - 6 + j], W from constexpr ROData tables.
  // No clamps: GUARD rows absorb s<0; top over-reads (halo-only tile) land inside
  // sm (A1 region) and influence only discarded coefficients.
  {
    const int lane = tid & 31;
    const int wid  = tid >> 5;
    const int mi   = lane & 15;          // M row (A) / N col (B,D)
    const int kb   = (lane >> 4) * 2;    // K pair selector per ISA A/B layout

    // loop-invariant A fragments: one cached 8B load each (2560B table, L1-resident)
    v2f alo[10], ahi[10];
#pragma unroll
    for (int kt = 0; kt < 10; ++kt) {
      f2 a = TAB_LO.v[kt][lane];
      f2 h = TAB_HI.v[kt][lane];
      alo[kt].x = a.x;  alo[kt].y = a.y;
      ahi[kt].x = h.x;  ahi[kt].y = h.y;
    }

    for (int mt = wid; mt < (M1 + 15) / 16; mt += NT / 32) {  // 9 tiles over 8 waves
      const int m0 = mt * 16;
      // common base row: s0(kt=0,p=0) = 2*m0 - 6 + kb  (guard-shifted, >= 2)
      const int baseRow = OFF_IN + GUARD + 2 * m0 - 6 + kb;
      v8f dlo = {0.f,0.f,0.f,0.f,0.f,0.f,0.f,0.f};
      v8f dhi = {0.f,0.f,0.f,0.f,0.f,0.f,0.f,0.f};
#pragma unroll
      for (int kt = 0; kt < 10; ++kt) {
        v2f bf;
        bf.x = sm[(baseRow + kt * 4    ) * COLS + mi];  // base + kt*256B
        bf.y = sm[(baseRow + kt * 4 + 1) * COLS + mi];  // base + kt*256B + 64B
        dlo = __builtin_amdgcn_wmma_f32_16x16x4_f32(false, alo[kt], false, bf,
                                                    (short)0, dlo, false, false);
        dhi = __builtin_amdgcn_wmma_f32_16x16x4_f32(false, ahi[kt], false, bf,
                                                    (short)0, dhi, false, false);
      }
#pragma unroll
      for (int j = 0; j < 8; ++j) {
        int m = m0 + j + (lane >> 4) * 8;    // C/D layout: vgpr j -> M=j (+8 for hi lanes)
        if (m < M1) {
          sm[(OFF_A1 + m) * COLS + mi] = dlo[j];
          sm[(OFF_D1 + m) * COLS + mi] = dhi[j];
        }
      }
    }
  }
  __syncthreads();

  // ---- Stage 2: level-2 analysis (VALU, flat-parallel)
  for (int idx = tid; idx < M2 * COLS; idx += NT) {
    int i = idx >> 4, c = idx & 15;
    float sa = 0.f, sd = 0.f;
#pragma unroll
    for (int k = 0; k < 8; ++k) {
      int s = reflect(2 * i - 6 + k, M1);
      float v = sm[(OFF_A1 + s) * COLS + c];
      sa += F_GLO[k] * v;  sd += F_GHI[k] * v;
    }
    sm[OFF_A2 * COLS + idx] = sa;
    sm[OFF_D2 * COLS + idx] = sd;
  }
  __syncthreads();

  // ---- Stage 3: level-3 analysis
  for (int idx = tid; idx < M3 * COLS; idx += NT) {
    int i = idx >> 4, c = idx & 15;
    float sa = 0.f, sd = 0.f;
#pragma unroll
    for (int k = 0; k < 8; ++k) {
      int s = reflect(2 * i - 6 + k, M2);
      float v = sm[(OFF_A2 + s) * COLS + c];
      sa += F_GLO[k] * v;  sd += F_GHI[k] * v;
    }
    sm[OFF_A3 * COLS + idx] = sa;
    sm[OFF_D3 * COLS + idx] = sd;
  }
  __syncthreads();

  // ---- Stage 4: per-head diagonal scaling
  for (int idx = tid; idx < M3 * COLS; idx += NT) {
    int c = idx & 15;
    sm[OFF_A3 * COLS + idx] *= wA[h0 + c];
    sm[OFF_D3 * COLS + idx] *= wD[2 * H + h0 + c];
  }
  for (int idx = tid; idx < M2 * COLS; idx += NT) {
    int c = idx & 15;
    sm[OFF_D2 * COLS + idx] *= wD[1 * H + h0 + c];
  }
  for (int idx = tid; idx < M1 * COLS; idx += NT) {
    int c = idx & 15;
    sm[OFF_D1 * COLS + idx] *= wD[0 * H + h0 + c];
  }
  __syncthreads();

  // ---- Stage 5: synth level 3 -> r3 (into A2 slot)
  for (int idx = tid; idx < R3L * COLS; idx += NT) {
    int t = idx >> 4, c = idx & 15;
    int kp = 1 - (t & 1);
    float r = 0.f;
#pragma unroll
    for (int j = 0; j < 4; ++j) {
      int k = kp + 2 * j;
      int q = (t + k - 1) >> 1;
      r += F_SLO[k] * sm[(OFF_A3 + q) * COLS + c]
         + F_SHI[k] * sm[(OFF_D3 + q) * COLS + c];
    }
    sm[OFF_A2 * COLS + idx] = r;
  }
  __syncthreads();

  // ---- Stage 6: synth level 2 -> r2 (into A1 slot); a = r3[:M2], d = d2
  for (int idx = tid; idx < R2L * COLS; idx += NT) {
    int t = idx >> 4, c = idx & 15;
    int kp = 1 - (t & 1);
    float r = 0.f;
#pragma unroll
    for (int j = 0; j < 4; ++j) {
      int k = kp + 2 * j;
      int q = (t + k - 1) >> 1;
      r += F_SLO[k] * sm[(OFF_A2 + q) * COLS + c]
         + F_SHI[k] * sm[(OFF_D2 + q) * COLS + c];
    }
    sm[OFF_A1 * COLS + idx] = r;
  }
  __syncthreads();

  // ---- Stage 7: synth level 1 -> r1; a = r2[:M1], d = d1
  for (int idx = tid; idx < R1L * COLS; idx += NT) {
    int t = idx >> 4, c = idx & 15;
    int kp = 1 - (t & 1);
    float r = 0.f;
#pragma unroll
    for (int j = 0; j < 4; ++j) {
      int k = kp + 2 * j;
      int q = (t + k - 1) >> 1;
      r += F_SLO[k] * sm[(OFF_A1 + q) * COLS + c]
         + F_SHI[k] * sm[(OFF_D1 + q) * COLS + c];
    }
    sm[OFF_R1 * COLS + idx] = r;
  }
  __syncthreads();

  // ---- Stage 8: out = r1[HALO..HALO+CHUNK) + x  (float4 stores, coalesced in h)
#pragma unroll
  for (int t = 0; t < (CHUNK * COLS) / 4 / NT; ++t) {   // 2 iters
    int idx4 = tid + t * NT;
    int i  = idx4 >> 2;
    int c4 = (idx4 & 3) * 4;
    const float4 r  = *(const float4*)&sm[(OFF_R1 + HALO + i) * COLS + c4];
    const float4 xi = *(const float4*)&sm[(OFF_IN + GUARD + HALO + i) * COLS + c4];
    float4 o;
    o.x = r.x + xi.x;  o.y = r.y + xi.y;  o.z = r.z + xi.z;  o.w = r.w + xi.w;
    *(float4*)(out + ((size_t)b * S + chunk * CHUNK + i) * H + h0 + c4) = o;
  }
}

extern "C" void kernel_launch(void* const* d_in, const int* in_sizes, int n_in,
                              void* d_out, int out_size, void* d_ws, size_t ws_size,
                              hipStream_t stream) {
  (void)in_sizes; (void)n_in; (void)out_size; (void)d_ws; (void)ws_size;
  const float* x  = (const float*)d_in[0];
  const float* wA = (const float*)d_in[1];   // (768,)
  const float* wD = (const float*)d_in[2];   // (3, 768)
  float* out = (float*)d_out;

  dim3 grid(S / CHUNK, H / COLS, NB);        // 32 x 48 x 8
  dim3 block(NT);
  wavelet_mix_kernel<<<grid, block, 0, stream>>>(x, wA, wD, out);
}